// SkipGramModel_82214263980494
// MI455X (gfx1250) — compile-verified
//
#include <hip/hip_runtime.h>
#include <hip/hip_bf16.h>

typedef __attribute__((ext_vector_type(2))) float v2f;
typedef __attribute__((ext_vector_type(4))) float v4f;
typedef __attribute__((ext_vector_type(8))) float v8f;

#define WAVES 4
#define LSTR  132   // 128 floats + 4-float pad -> conflict-free column reads

__device__ __forceinline__ float log_sigmoid(float x) {
    // log sigmoid(x) = min(x,0) - log1p(exp(-|x|))  (numerically stable)
    return fminf(x, 0.0f) - log1pf(expf(-fabsf(x)));
}

__global__ __launch_bounds__(128) void sg_loss_kernel(
    const int*   __restrict__ targets,   // [N]
    const int*   __restrict__ contexts,  // [N]
    const int*   __restrict__ negs,      // [N*5]
    const float* __restrict__ Wh,        // [V*128]
    const float* __restrict__ Wo,        // [V*128]
    float*       __restrict__ partials,  // [gridDim.x]
    int numTiles)
{
    __shared__ float At[WAVES][16 * LSTR];
    __shared__ float Bt[WAVES][16 * LSTR];
    __shared__ float dP[WAVES][16];
    __shared__ float dN[WAVES][16];
    __shared__ float wsum[WAVES];

    const int lane = threadIdx.x & 31;
    const int wid  = threadIdx.x >> 5;
    const int m    = lane & 15;   // matrix row (A) / col (B) this lane feeds
    const int hi   = lane >> 4;   // half-wave selector in WMMA operand layout

    float waveAcc = 0.0f;

    const int waveGlobal = blockIdx.x * WAVES + wid;
    const int waveStride = gridDim.x * WAVES;

    for (int tile = waveGlobal; tile < numTiles; tile += waveStride) {
        const int pbase = tile * 16;

        // ---- stage T (A) and C (B) tiles: one 512B row per wave-load, coalesced ----
        #pragma unroll
        for (int r = 0; r < 16; ++r) {
            const int ti = targets[pbase + r];    // uniform -> scalar load
            const int ci = contexts[pbase + r];   // uniform -> scalar load
            v4f tv = *(const v4f*)(Wh + (size_t)ti * 128 + lane * 4);
            v4f cv = *(const v4f*)(Wo + (size_t)ci * 128 + lane * 4);
            *(v4f*)&At[wid][r * LSTR + lane * 4] = tv;
            *(v4f*)&Bt[wid][r * LSTR + lane * 4] = cv;
        }

        // ---- pos scores: D = T x C^T via f32 WMMA, K=128 in chunks of 4 ----
        v8f acc = {};
        #pragma unroll
        for (int kk = 0; kk < 128; kk += 4) {
            v2f a = *(const v2f*)&At[wid][m * LSTR + kk + 2 * hi];
            v2f b;
            b.x = Bt[wid][m * LSTR + kk + hi];
            b.y = Bt[wid][m * LSTR + kk + hi + 2];
            acc = __builtin_amdgcn_wmma_f32_16x16x4_f32(false, a, false, b,
                                                        (short)0, acc, false, false);
        }
        {   // diag(i,i): VGPR i%8, lane i (i<8) or lane i+16 (i>=8)
            const int j = lane & 7;
            float dval = acc[0];
            #pragma unroll
            for (int q = 1; q < 8; ++q) dval = (j == q) ? acc[q] : dval;
            if (lane < 8)   dP[wid][lane]      = dval;
            if (lane >= 24) dP[wid][lane - 16] = dval;
        }

        // ---- stage Nsum tile into B: elementwise sum of the 5 negative rows ----
        #pragma unroll
        for (int r = 0; r < 16; ++r) {
            const int nb = (pbase + r) * 5;
            v4f s = {};
            #pragma unroll
            for (int k = 0; k < 5; ++k) {
                const int ni = negs[nb + k];      // uniform -> scalar load
                v4f nv = *(const v4f*)(Wo + (size_t)ni * 128 + lane * 4);
                s.x += nv.x; s.y += nv.y; s.z += nv.z; s.w += nv.w;
            }
            *(v4f*)&Bt[wid][r * LSTR + lane * 4] = s;
        }

        // ---- neg scores: D = T x Nsum^T ----
        v8f acc2 = {};
        #pragma unroll
        for (int kk = 0; kk < 128; kk += 4) {
            v2f a = *(const v2f*)&At[wid][m * LSTR + kk + 2 * hi];
            v2f b;
            b.x = Bt[wid][m * LSTR + kk + hi];
            b.y = Bt[wid][m * LSTR + kk + hi + 2];
            acc2 = __builtin_amdgcn_wmma_f32_16x16x4_f32(false, a, false, b,
                                                         (short)0, acc2, false, false);
        }
        {
            const int j = lane & 7;
            float dval = acc2[0];
            #pragma unroll
            for (int q = 1; q < 8; ++q) dval = (j == q) ? acc2[q] : dval;
            if (lane < 8)   dN[wid][lane]      = dval;
            if (lane >= 24) dN[wid][lane - 16] = dval;
        }

        // ---- per-pair loss for the 16 pairs of this tile ----
        float v = 0.0f;
        if (lane < 16) {
            v = log_sigmoid(dP[wid][lane]) + log_sigmoid(-dN[wid][lane]);
        }
        waveAcc += v;
    }

    // fixed-order wave reduction (wave32)
    #pragma unroll
    for (int off = 16; off >= 1; off >>= 1)
        waveAcc += __shfl_xor(waveAcc, off, 32);
    if (lane == 0) wsum[wid] = waveAcc;
    __syncthreads();
    if (threadIdx.x == 0) {
        float b = 0.0f;
        #pragma unroll
        for (int w = 0; w < WAVES; ++w) b += wsum[w];
        partials[blockIdx.x] = b;
    }
}

__global__ __launch_bounds__(256) void sg_reduce_kernel(
    const float* __restrict__ partials, int n,
    float* __restrict__ out, float invN)
{
    __shared__ float s[256];
    float a = 0.0f;
    for (int i = threadIdx.x; i < n; i += 256) a += partials[i];
    s[threadIdx.x] = a;
    __syncthreads();
    for (int o = 128; o > 0; o >>= 1) {
        if ((int)threadIdx.x < o) s[threadIdx.x] += s[threadIdx.x + o];
        __syncthreads();
    }
    if (threadIdx.x == 0) out[0] = -s[0] * invN;
}

extern "C" void kernel_launch(void* const* d_in, const int* in_sizes, int n_in,
                              void* d_out, int out_size, void* d_ws, size_t ws_size,
                              hipStream_t stream) {
    const int*   targets  = (const int*)d_in[0];
    const int*   contexts = (const int*)d_in[1];
    const int*   negs     = (const int*)d_in[2];
    const float* Wh       = (const float*)d_in[3];
    const float* Wo       = (const float*)d_in[4];
    float*       out      = (float*)d_out;
    float*       partials = (float*)d_ws;

    const int N        = in_sizes[0];   // 524288, divisible by 16
    const int numTiles = N / 16;
    const int blocks   = 1024;

    sg_loss_kernel<<<blocks, 128, 0, stream>>>(targets, contexts, negs, Wh, Wo,
                                               partials, numTiles);
    sg_reduce_kernel<<<1, 256, 0, stream>>>(partials, blocks, out, 1.0f / (float)N);
}